// AdjustHueSaturation_60455959658579
// MI455X (gfx1250) — compile-verified
//
#include <hip/hip_runtime.h>

// (B, C, H, W) = (64, 3, 512, 512), fp32, planar layout.
#define HWSZ (512 * 512)
#define NBATCH 64
#define IPT 2                       // float4 groups per thread

typedef __attribute__((ext_vector_type(4))) float v4f;

__device__ __forceinline__ float clamp01(float x) {
    return fminf(fmaxf(x, 0.0f), 1.0f);
}

// channel_k = v - c * clamp01(min(n, 4-n)), n = mod(k + H6, 6), k+H6 in [k, k+6)
__device__ __forceinline__ float chan(float n, float v, float c) {
    n = (n >= 6.0f) ? (n - 6.0f) : n;
    float x = clamp01(fminf(n, 4.0f - n));
    return fmaf(-c, x, v);
}

// One pixel: RGB -> HSV -> (hue shift, sat scale) -> RGB, closed form.
__device__ __forceinline__ void adjust_px(float r, float g, float b,
                                          float dh6, float dsat,
                                          float& ro, float& go, float& bo) {
    float maxc = fmaxf(r, fmaxf(g, b));   // v_max3_num_f32
    float minc = fminf(r, fminf(g, b));   // v_min3_num_f32
    float cr   = maxc - minc;
    bool  eq   = (maxc == minc);

    // s = cr / (eq ? 1 : maxc);  rc/gc/bc share 1/(eq ? 1 : cr)
    float s   = cr * __builtin_amdgcn_rcpf(eq ? 1.0f : maxc);
    float inv = __builtin_amdgcn_rcpf(eq ? 1.0f : cr);
    float rc = (maxc - r) * inv;
    float gc = (maxc - g) * inv;
    float bc = (maxc - b) * inv;

    // hr/hg/hb are mutually exclusive -> single nested select; hsum in [-1, 5]
    float hsum = (maxc == r) ? (bc - gc)
               : (maxc == g) ? (2.0f + rc - bc)
               :               (4.0f + gc - rc);

    // H6 = 6 * mod(mod(hsum,6)/6 + dh, 1) == mod(hsum + 6*dh, 6); arg in [-4, 8)
    float ha = hsum + dh6;
    float H6 = fmaf(-6.0f, floorf(ha * (1.0f / 6.0f)), ha);   // in [0, 6) (+-1ulp)

    float ss = clamp01(s * dsat);   // mod_sat
    float v  = maxc;
    float c  = v * ss;              // chroma after sat scale

    // R/G/B tables [v,q,p,p,t,v] / [t,v,v,q,p,p] / [p,p,t,v,v,q] == chan at k=5/3/1
    ro = chan(5.0f + H6, v, c);
    go = chan(3.0f + H6, v, c);
    bo = chan(1.0f + H6, v, c);
}

__global__ __launch_bounds__(256) void
AdjustHueSaturation_60455959658579_kernel(const float* __restrict__ img,
                                          const float* __restrict__ xform,
                                          float* __restrict__ out) {
    const int batch = blockIdx.y;
    // Block covers 256*4*IPT = 2048 contiguous floats per plane.
    const int base0 = batch * 3 * HWSZ + blockIdx.x * (256 * 4 * IPT) + threadIdx.x * 4;

    // Uniform per-block -> scalar path; prescale hue delta by 6.
    const float dh6 = 6.0f * xform[2 * batch + 0];
    const float ds  = xform[2 * batch + 1];

    // Issue all NT b128 loads first (one clause, 6 outstanding).
    v4f R[IPT], G[IPT], Bl[IPT];
#pragma unroll
    for (int j = 0; j < IPT; ++j) {
        const int base = base0 + j * 1024;                    // +4KB instr offset
        R[j]  = __builtin_nontemporal_load(reinterpret_cast<const v4f*>(img + base));
        G[j]  = __builtin_nontemporal_load(reinterpret_cast<const v4f*>(img + base + HWSZ));
        Bl[j] = __builtin_nontemporal_load(reinterpret_cast<const v4f*>(img + base + 2 * HWSZ));
    }

#pragma unroll
    for (int j = 0; j < IPT; ++j) {
        v4f Ro, Go, Bo;
#pragma unroll
        for (int k = 0; k < 4; ++k) {
            float ro, go, bo;
            adjust_px(R[j][k], G[j][k], Bl[j][k], dh6, ds, ro, go, bo);
            Ro[k] = ro;
            Go[k] = go;
            Bo[k] = bo;
        }
        const int base = base0 + j * 1024;
        __builtin_nontemporal_store(Ro, reinterpret_cast<v4f*>(out + base));
        __builtin_nontemporal_store(Go, reinterpret_cast<v4f*>(out + base + HWSZ));
        __builtin_nontemporal_store(Bo, reinterpret_cast<v4f*>(out + base + 2 * HWSZ));
    }
}

extern "C" void kernel_launch(void* const* d_in, const int* in_sizes, int n_in,
                              void* d_out, int out_size, void* d_ws, size_t ws_size,
                              hipStream_t stream) {
    (void)in_sizes; (void)n_in; (void)out_size; (void)d_ws; (void)ws_size;
    const float* img   = (const float*)d_in[0];   // (64, 3, 512, 512) fp32
    const float* xform = (const float*)d_in[1];   // (64, 2) fp32: [dh, ds]
    float*       out   = (float*)d_out;

    // HWSZ / (256*4*IPT) = 128 blocks per image, exact coverage, no tail.
    dim3 block(256, 1, 1);
    dim3 grid(HWSZ / (256 * 4 * IPT), NBATCH, 1);
    AdjustHueSaturation_60455959658579_kernel<<<grid, block, 0, stream>>>(img, xform, out);
}